// OIMLoss_31069793419670
// MI455X (gfx1250) — compile-verified
//
#include <hip/hip_runtime.h>
#include <math.h>

#define BATCH   1024
#define FEAT    128
#define NCLS    100000
#define MARGIN  0.1f
#define SCALAR  1.0f
#define EPSF    1e-12f

typedef __attribute__((ext_vector_type(2))) float v2f;
typedef __attribute__((ext_vector_type(4))) float v4f;
typedef __attribute__((ext_vector_type(8))) float v8f;

// ---------------------------------------------------------------------------
// 1) L2-normalize the 1024 feature rows. One wave32 per row, 4 floats/lane.
// ---------------------------------------------------------------------------
__global__ void oim_normalize(const float* __restrict__ in, float* __restrict__ xn) {
    int wave = (blockIdx.x * blockDim.x + threadIdx.x) >> 5;
    int lane = threadIdx.x & 31;
    if (wave >= BATCH) return;
    const float* row = in + (size_t)wave * FEAT;
    v4f v = *(const v4f*)(row + lane * 4);
    float ss = v.x * v.x + v.y * v.y + v.z * v.z + v.w * v.w;
#pragma unroll
    for (int off = 16; off > 0; off >>= 1)
        ss += __shfl_xor(ss, off, 32);
    float inv = 1.0f / (sqrtf(ss) + EPSF);
    v4f o = { v.x * inv, v.y * inv, v.z * inv, v.w * inv };
    *(v4f*)(xn + (size_t)wave * FEAT + lane * 4) = o;
}

// ---------------------------------------------------------------------------
// 2) logits = x_hat @ lut^T via V_WMMA_F32_16X16X4_F32.
//    Block = 256 threads = 8 waves; block covers 128 columns of lut.
//    Each wave owns 16 columns: loads its 32 B-fragments once (held in VGPRs),
//    then loops the 64 M-tiles (A-frags from L2-resident x_hat), 32 WMMA/tile.
//    A frag  (16x4, f32): lane[0..15] -> M row, half 0 holds K,K+1 / half 1 K+2,K+3
//    B frag  (4x16, f32): lane[0..15] -> N col, half split over K the same way
//    C tile  (16x16,f32): VGPR j -> M = j + 8*half, lane[0..15] -> N
// ---------------------------------------------------------------------------
__global__ void oim_gemm(const float* __restrict__ xn, const float* __restrict__ lut,
                         float* __restrict__ logits) {
    int wave = threadIdx.x >> 5;
    int lane = threadIdx.x & 31;
    int col0 = blockIdx.x * 128 + wave * 16;
    if (col0 >= NCLS) return;                 // wave-uniform guard (EXEC all-1 inside)
    int l15  = lane & 15;
    int half = lane >> 4;

    // Preload B fragments for this wave's 16 lut rows across all of K=128.
    v2f bfr[32];
    const float* bp = lut + (size_t)(col0 + l15) * FEAT + half * 2;
#pragma unroll
    for (int ks = 0; ks < 32; ++ks)
        bfr[ks] = *(const v2f*)(bp + ks * 4);

#pragma unroll 1
    for (int mt = 0; mt < BATCH / 16; ++mt) {
        const float* ap = xn + (size_t)(mt * 16 + l15) * FEAT + half * 2;
        v8f acc = {};
#pragma unroll
        for (int ks = 0; ks < 32; ++ks) {
            v2f a = *(const v2f*)(ap + ks * 4);
            acc = __builtin_amdgcn_wmma_f32_16x16x4_f32(
                /*neg_a=*/false, a, /*neg_b=*/false, bfr[ks],
                /*c_mod=*/(short)0, acc, /*reuse_a=*/false, /*reuse_b=*/false);
        }
        // Stream the 16x16 tile out non-temporally (logits >> L2 capacity).
        float* outp = logits + (size_t)(mt * 16 + half * 8) * NCLS + (col0 + l15);
#pragma unroll
        for (int j = 0; j < 8; ++j) {
            __builtin_nontemporal_store(acc[j] * SCALAR, outp);
            outp += NCLS;
        }
    }
}

// ---------------------------------------------------------------------------
// 3) Additive margin on the target entry of each row (post-scale amount).
// ---------------------------------------------------------------------------
__global__ void oim_margin(const int* __restrict__ tgt, float* __restrict__ logits) {
    int r = blockIdx.x * blockDim.x + threadIdx.x;
    if (r < BATCH) {
        size_t idx = (size_t)r * NCLS + (size_t)tgt[r];
        logits[idx] -= MARGIN * SCALAR;
    }
}

// ---------------------------------------------------------------------------
// 4) Per-row (max, sum-exp) in one streaming pass: online softmax per thread,
//    then deterministic LDS log-sum-exp tree merge. One block per row.
// ---------------------------------------------------------------------------
__global__ void oim_rowstats(const float* __restrict__ logits,
                             float* __restrict__ mrow, float* __restrict__ srow) {
    __shared__ float sm[256];
    __shared__ float ss[256];
    int r = blockIdx.x;
    const float* row = logits + (size_t)r * NCLS;
    float m = -3.402823466e38f, s = 0.0f;
    for (int c = threadIdx.x; c < NCLS; c += 256) {
        float v = __builtin_nontemporal_load(row + c);
        if (v > m) { s = s * __expf(m - v) + 1.0f; m = v; }
        else       { s += __expf(v - m); }
    }
    sm[threadIdx.x] = m;
    ss[threadIdx.x] = s;
    __syncthreads();
#pragma unroll
    for (int off = 128; off > 0; off >>= 1) {
        if (threadIdx.x < off) {
            float m1 = sm[threadIdx.x], s1 = ss[threadIdx.x];
            float m2 = sm[threadIdx.x + off], s2 = ss[threadIdx.x + off];
            float mm = fmaxf(m1, m2);
            sm[threadIdx.x] = mm;
            ss[threadIdx.x] = s1 * __expf(m1 - mm) + s2 * __expf(m2 - mm);
        }
        __syncthreads();
    }
    if (threadIdx.x == 0) { mrow[r] = sm[0]; srow[r] = ss[0]; }
}

// ---------------------------------------------------------------------------
// 5) loss = mean over rows of (m + log s - logit_target). One 1024-thread block.
// ---------------------------------------------------------------------------
__global__ void oim_loss(const int* __restrict__ tgt, const float* __restrict__ logits,
                         const float* __restrict__ mrow, const float* __restrict__ srow,
                         float* __restrict__ out) {
    __shared__ float red[BATCH];
    int r = threadIdx.x;
    float lt = logits[(size_t)r * NCLS + (size_t)tgt[r]];
    red[r] = mrow[r] + __logf(srow[r]) - lt;
    __syncthreads();
#pragma unroll
    for (int off = 512; off > 0; off >>= 1) {
        if (r < off) red[r] += red[r + off];
        __syncthreads();
    }
    if (r == 0) out[0] = red[0] / (float)BATCH;
}

// ---------------------------------------------------------------------------
extern "C" void kernel_launch(void* const* d_in, const int* in_sizes, int n_in,
                              void* d_out, int out_size, void* d_ws, size_t ws_size,
                              hipStream_t stream) {
    const float* inputs  = (const float*)d_in[0];
    const int*   targets = (const int*)d_in[1];
    const float* lut     = (const float*)d_in[2];

    float* out    = (float*)d_out;
    float* logits = out + 1;                 // d_out = [loss, logits(1024x100000)]

    float* xn   = (float*)d_ws;              // 1024*128 f32 = 512 KB
    float* mrow = xn + BATCH * FEAT;         // 1024 f32
    float* srow = mrow + BATCH;              // 1024 f32

    oim_normalize<<<BATCH / 8, 256, 0, stream>>>(inputs, xn);
    oim_gemm<<<(NCLS + 127) / 128, 256, 0, stream>>>(xn, lut, logits);
    oim_margin<<<(BATCH + 255) / 256, 256, 0, stream>>>(targets, logits);
    oim_rowstats<<<BATCH, 256, 0, stream>>>(logits, mrow, srow);
    oim_loss<<<1, BATCH, 0, stream>>>(targets, logits, mrow, srow, out);
}